// ROIPool_85478439125450
// MI455X (gfx1250) — compile-verified
//
#include <hip/hip_runtime.h>
#include <math.h>

// ROI max-pool for MI455X (gfx1250, wave32).
// feat: (B=4, C=256, H=224, W=224) fp32, NCHW
// rois: (512, 5) fp32 = [batch, x1, y1, x2, y2]
// out : (512, 256, 7, 7) fp32
//
// Launch geometry does the index decomposition (no 64-bit div/mod):
//   grid  = (roi*256 + c, ph)   -> scalar (SGPR) per block
//   block = (32, 7), wave = threadIdx.y = pw (wave-uniform, readfirstlane'd)
// One wave32 per output bin. Lanes stride the bin's flattened (h,w) region so
// consecutive lanes read consecutive w (coalesced in NCHW). dh = i/bw via
// round-up magic (exact for bw<2^7, i<2^25). global_prefetch_b8 pre-touches
// each region row; wave32 shuffle-xor max reduction (5 steps).

namespace {
constexpr int kC = 256;
constexpr int kH = 224;
constexpr int kW = 224;
constexpr int kNRois = 512;
constexpr int kPool = 7;
constexpr float kScale = 0.0625f;
constexpr int kBins = kPool * kPool;  // 49
}  // namespace

// floor(x*scale + 0.5) with fp-contraction blocked (reference does mul then
// add as separate fp32 ops; an fma would change rounding).
__device__ __forceinline__ int round_half_up_scaled(float v) {
  return (int)floorf(__fadd_rn(__fmul_rn(v, kScale), 0.5f));
}

__global__ __launch_bounds__(224) void ROIPool_85478439125450_kernel(
    const float* __restrict__ feat, const float* __restrict__ rois,
    float* __restrict__ out) {
  const int lane = threadIdx.x;                                  // 0..31
  const int pw = __builtin_amdgcn_readfirstlane(threadIdx.y);    // wave-uniform
  const int rc = blockIdx.x;      // roi*256 + c   (scalar)
  const int ph = blockIdx.y;      // scalar
  const int c = rc & (kC - 1);
  const int roi = rc >> 8;

  // ROI decode: scalar offset -> constant-cache s_load path.
  const float* __restrict__ r = rois + roi * 5;
  const int b  = (int)r[0];
  const int xs = round_half_up_scaled(r[1]);
  const int ys = round_half_up_scaled(r[2]);
  const int xe = round_half_up_scaled(r[3]);
  const int ye = round_half_up_scaled(r[4]);

  const int roi_w = max(xe - xs + 1, 1);
  const int roi_h = max(ye - ys + 1, 1);
  const float bin_w = (float)roi_w / (float)kPool;  // IEEE fp32 divide
  const float bin_h = (float)roi_h / (float)kPool;

  const int hs = min(max((int)floorf(__fmul_rn((float)ph, bin_h)) + ys, 0), kH);
  const int he =
      min(max((int)ceilf(__fmul_rn((float)(ph + 1), bin_h)) + ys, 0), kH);
  const int ws = min(max((int)floorf(__fmul_rn((float)pw, bin_w)) + xs, 0), kW);
  const int we =
      min(max((int)ceilf(__fmul_rn((float)(pw + 1), bin_w)) + xs, 0), kW);

  const int bh = he - hs;
  const int bw = we - ws;
  const bool valid = (bh > 0) && (bw > 0);

  // Bin-origin pointer (wave-uniform): feat[b][c][hs][ws]
  const float* __restrict__ fp =
      feat + (((long long)(b * kC + c) * kH + hs) * kW + ws);

  float m = -INFINITY;
  if (valid) {
    // Pre-touch each region row's first cacheline (global_prefetch_b8).
    for (int hh = lane; hh < bh; hh += 32) {
      __builtin_prefetch(fp + hh * kW, 0, 0);
    }

    // Round-up magic divisor for dh = i / bw (bw is wave-uniform scalar).
    // Exact for bw < 2^7 and i < 2^25 (here bw <= ~35, n <= ~1200).
    const unsigned mgk =
        (bw > 1) ? (unsigned)(0x100000000ULL / (unsigned)bw) + 1u : 0u;

    const int n = bh * bw;
    for (int i = lane; i < n; i += 32) {
      const int dh = (bw > 1) ? (int)__umulhi((unsigned)i, mgk) : i;
      const int dw = i - dh * bw;
      m = fmaxf(m, fp[dh * kW + dw]);
    }
  }

  // Wave32 max reduction (CDNA5 is wave32-only: 5 xor steps).
#pragma unroll
  for (int off = 16; off > 0; off >>= 1) {
    m = fmaxf(m, __shfl_xor(m, off, 32));
  }

  if (lane == 0) {
    out[rc * kBins + ph * kPool + pw] = valid ? m : 0.0f;
  }
}

extern "C" void kernel_launch(void* const* d_in, const int* in_sizes, int n_in,
                              void* d_out, int out_size, void* d_ws,
                              size_t ws_size, hipStream_t stream) {
  (void)in_sizes; (void)n_in; (void)d_ws; (void)ws_size; (void)out_size;
  const float* feat = (const float*)d_in[0];
  const float* rois = (const float*)d_in[1];
  float* out = (float*)d_out;

  dim3 grid(kNRois * kC, kPool);  // (131072, 7)
  dim3 block(32, kPool);          // 7 waves, one per pw
  hipLaunchKernelGGL(ROIPool_85478439125450_kernel, grid, block, 0, stream,
                     feat, rois, out);
}